// Funct2_33990371181223
// MI455X (gfx1250) — compile-verified
//
#include <hip/hip_runtime.h>

// -------------------------------------------------------------------------
// Lattice nearest-neighbor features + pointwise MLP(3->32 tanh ->1), summed
// per config. Layer-1 GEMM via V_WMMA_F32_16X16X4_F32 (K=3 padded to 4,
// exact fp32). Layer-2 + lattice sum collapse into a per-lane weighted
// accumulator (W2[h] depends only on lane). Branchless fragment setup so the
// inner loop has no EXEC divergence (WMMA requires EXEC all-ones anyway).
// -------------------------------------------------------------------------

typedef __attribute__((ext_vector_type(2))) float v2f;
typedef __attribute__((ext_vector_type(8))) float v8f;

#define LSIZE 512
#define LMASK 511

__device__ __forceinline__ float fast_tanh(float v) {
#if __has_builtin(__builtin_amdgcn_tanhf)
    return __builtin_amdgcn_tanhf(v);          // gfx1250 v_tanh_f32
#else
    float e = __expf(2.0f * v);                // v_exp_f32 path
    return 1.0f - 2.0f / (e + 1.0f);
#endif
}

__global__ void zero_out_kernel(float* __restrict__ out, int n) {
    int i = threadIdx.x + blockIdx.x * blockDim.x;
    if (i < n) out[i] = 0.0f;
}

__global__ void __launch_bounds__(256)
lattice_mlp_wmma_kernel(const float* __restrict__ x,
                        const float* __restrict__ W1,   // (32,3) row-major
                        const float* __restrict__ b1,   // (32,)
                        const float* __restrict__ W2,   // (1,32)
                        const float* __restrict__ b2,   // (1,)
                        float* __restrict__ out) {      // (B,)
    __shared__ float red[8];

    const int row  = blockIdx.x;            // 0 .. B*512-1
    const int b    = row >> 9;              // config index
    const int l    = row & LMASK;           // lattice row
    const int lane = threadIdx.x & 31;
    const int wave = threadIdx.x >> 5;      // 0..7
    const int n    = lane & 15;             // N index (h % 16) / site sub-index
    const bool hiK = (lane >= 16);

    // ---- B matrices (4x16, K x N) for the two hidden halves -------------
    // Assumed 32-bit B layout (mirrors documented A layout):
    //   VGPR0: lanes 0-15 -> K=0, lanes 16-31 -> K=2
    //   VGPR1: lanes 0-15 -> K=1, lanes 16-31 -> K=3
    // B[k][h] = W1[h][k], K=3 column is zero padding.  Branchless selects.
    const float w1_0lo = W1[n * 3 + 0],        w1_1lo = W1[n * 3 + 1];
    const float w1_2lo = W1[n * 3 + 2];
    const float w1_0hi = W1[(n + 16) * 3 + 0], w1_1hi = W1[(n + 16) * 3 + 1];
    const float w1_2hi = W1[(n + 16) * 3 + 2];

    v2f Blo, Bhi;
    Blo.x = hiK ? w1_2lo : w1_0lo;   Blo.y = hiK ? 0.0f : w1_1lo;
    Bhi.x = hiK ? w1_2hi : w1_0hi;   Bhi.y = hiK ? 0.0f : w1_1hi;

    const float w2lo = W2[n];
    const float w2hi = W2[n + 16];

    // C accumulator preloaded with bias b1[h]  (h = N = lane&15 (+16))
    const float b1lo = b1[n];
    const float b1hi = b1[n + 16];
    v8f Clo, Chi;
#pragma unroll
    for (int r = 0; r < 8; ++r) { Clo[r] = b1lo; Chi[r] = b1hi; }

    const float* __restrict__ xb   = x + (size_t)b * (LSIZE * LSIZE);
    const int lm1 = (l + LMASK) & LMASK;
    const int lp1 = (l + 1) & LMASK;
    const float* __restrict__ rowc = xb + l   * LSIZE;
    const float* __restrict__ rowu = xb + lm1 * LSIZE;
    const float* __restrict__ rowd = xb + lp1 * LSIZE;

    float acc = 0.0f;

    // Each wave covers 64 columns = 4 tiles of 16 sites.
#pragma unroll
    for (int j = 0; j < 4; ++j) {
        const int m   = wave * 64 + j * 16 + n;        // site column
        const int mm1 = (m + LMASK) & LMASK;
        const int mp1 = (m + 1) & LMASK;

        const float c  = rowc[m];
        const float sx = rowc[mm1] + rowc[mp1];        // horizontal neighbors
        const float x1 = sx + rowu[m] + rowd[m];       // 4-neighbor sum

        // Features: [x^2, x1*x, sx*x, 0]; branchless lane-select into the
        // documented 32-bit A layout:
        //   VGPR0: lanes 0-15 -> K=0 (x^2),  lanes 16-31 -> K=2 (sx*x)
        //   VGPR1: lanes 0-15 -> K=1 (x1*x), lanes 16-31 -> K=3 (0)
        const float f0 = c * c;
        const float f1 = x1 * c;
        const float f2 = sx * c;
        v2f A;
        A.x = hiK ? f2 : f0;
        A.y = hiK ? 0.0f : f1;

        v8f dlo = __builtin_amdgcn_wmma_f32_16x16x4_f32(
            false, A, false, Blo, (short)0, Clo, false, false);
        v8f dhi = __builtin_amdgcn_wmma_f32_16x16x4_f32(
            false, A, false, Bhi, (short)0, Chi, false, false);

        float slo = 0.0f, shi = 0.0f;
#pragma unroll
        for (int r = 0; r < 8; ++r) {
            slo += fast_tanh(dlo[r]);
            shi += fast_tanh(dhi[r]);
        }
        acc = fmaf(w2lo, slo, acc);
        acc = fmaf(w2hi, shi, acc);
    }

    // ---- wave32 butterfly reduction -------------------------------------
#pragma unroll
    for (int off = 16; off > 0; off >>= 1)
        acc += __shfl_xor(acc, off, 32);

    if (lane == 0) red[wave] = acc;
    __syncthreads();

    if (threadIdx.x == 0) {
        float s = 0.0f;
#pragma unroll
        for (int w = 0; w < 8; ++w) s += red[w];
        s += (float)LSIZE * b2[0];   // block covers 512 sites: + L*b2 per row
        atomicAdd(&out[b], s);
    }
}

extern "C" void kernel_launch(void* const* d_in, const int* in_sizes, int n_in,
                              void* d_out, int out_size, void* d_ws, size_t ws_size,
                              hipStream_t stream) {
    (void)n_in; (void)d_ws; (void)ws_size;
    const float* x  = (const float*)d_in[0];
    const float* W1 = (const float*)d_in[1];
    const float* b1 = (const float*)d_in[2];
    const float* W2 = (const float*)d_in[3];
    const float* b2 = (const float*)d_in[4];
    float* out = (float*)d_out;

    const int B = in_sizes[0] / (LSIZE * LSIZE);   // 16

    zero_out_kernel<<<1, 64, 0, stream>>>(out, out_size);
    lattice_mlp_wmma_kernel<<<B * LSIZE, 256, 0, stream>>>(x, W1, b1, W2, b2, out);
}